// Decoder_55336358642786
// MI455X (gfx1250) — compile-verified
//
#include <hip/hip_runtime.h>
#include <hip/hip_bf16.h>

// ---------------------------------------------------------------------------
// CDNA5 (gfx1250) LSTM decoder:  3-layer LSTM (B=16,H=1024,T=257) + attention
// + MLP + NLL, all GEMMs on V_WMMA_F32_16X16X4_F32 (wave32, 16x16 tiles).
// ---------------------------------------------------------------------------

typedef __attribute__((ext_vector_type(2))) float v2f;
typedef __attribute__((ext_vector_type(8))) float v8f;

#define BATCH 16
#define SEQL  256
#define TT    257          // T = L+1
#define EE    512          // embedding dim
#define HH    1024         // hidden dim (== ENC2)
#define SS    512          // encoder positions
#define VV    1024         // vocab

static __device__ __forceinline__ v8f wmma4(v2f a, v2f b, v8f c) {
  // D = A(16x4) * B(4x16) + C(16x16), all f32
  return __builtin_amdgcn_wmma_f32_16x16x4_f32(false, a, false, b, (short)0, c,
                                               false, false);
}

// Generic K-loop for one 16x16 tile with two independent accumulator chains
// (breaks the WMMA D->C dependency chain for pipe overlap).
// aP: per-lane &A[m][0] (element stride sa); bP: per-lane &B'[n][0] where
// B'[n][k] = B[k][n] (element stride sb). kh = lane>>4.
// Uniform (sgpr) trip count -> EXEC stays all-1s around v_wmma.  K % 8 == 0.
static __device__ __forceinline__ v8f gemm_tile(const float* __restrict__ aP, int sa,
                                                const float* __restrict__ bP, int sb,
                                                int K, int kh, v8f acc) {
  v8f acc2 = {0.f, 0.f, 0.f, 0.f, 0.f, 0.f, 0.f, 0.f};
#pragma unroll 4
  for (int k0 = 0; k0 < K; k0 += 8) {
    const int ka = k0 + 2 * kh;
    const int kb = ka + 4;
    v2f a0 = {aP[(size_t)ka * sa], aP[(size_t)(ka + 1) * sa]};
    v2f b0 = {bP[(size_t)ka * sb], bP[(size_t)(ka + 1) * sb]};
    acc = wmma4(a0, b0, acc);
    v2f a1 = {aP[(size_t)kb * sa], aP[(size_t)(kb + 1) * sa]};
    v2f b1 = {bP[(size_t)kb * sb], bP[(size_t)(kb + 1) * sb]};
    acc2 = wmma4(a1, b1, acc2);
  }
  return acc + acc2;
}

static __device__ __forceinline__ float sigmoidf_(float x) {
  return 1.f / (1.f + __expf(-x));
}

// ---------------------------------------------------------------------------
// One LSTM layer, one time step.  grid = 64 blocks (16 h-columns each),
// block = 256 threads = 8 waves; wave w owns gate (w&3) and K-half (w>>2).
// A-panel (16 x (Ka+H)) staged through LDS in 512-wide K chunks; each wave
// runs two independent WMMA accumulator chains over its 256-K half-chunk.
// K-half is folded into the base pointers so the WMMA loop has literal
// bounds (uniform control flow + deep load pipelining).
// ---------------------------------------------------------------------------
__global__ __launch_bounds__(256) void lstm_step_kernel(
    int t, int layer,
    const int* __restrict__ tokens, const float* __restrict__ emb,
    const float* __restrict__ W_ih, int ih_stride, int Ka,
    const float* __restrict__ W_hh,
    const float* __restrict__ b_ih, const float* __restrict__ b_hh,
    const float* __restrict__ h_below,   // layer>0: h of layer-1 at step t
    const float* __restrict__ h_prev,    // this layer's h at step t-1
    float* __restrict__ h_out,           // this layer's h at step t
    float* __restrict__ c_state,         // this layer's c (block-exclusive cols)
    float* __restrict__ hs)              // [B][T][H] output (layer 2 only)
{
  __shared__ float As[16 * 512];
  __shared__ float Gt[8 * 256];
  const int tid   = threadIdx.x;
  const int wave  = tid >> 5;
  const int lane  = tid & 31;
  const int kh    = lane >> 4;
  const int mn    = lane & 15;
  const int gate  = wave & 3;            // i,f,g,o
  const int kwave = wave >> 2;           // K-half within each staged chunk
  const int n0    = blockIdx.x * 16;     // h-column chunk base
  const int gcol  = gate * HH + n0;      // gate-column base for this wave
  const int kstart = kwave * 256;        // this wave's half of each chunk

  v8f acc0 = {0.f, 0.f, 0.f, 0.f, 0.f, 0.f, 0.f, 0.f};
  v8f acc1 = {0.f, 0.f, 0.f, 0.f, 0.f, 0.f, 0.f, 0.f};

  const int nch = (Ka + HH) >> 9;        // 3 (layer0) or 4 (layers 1,2)
  for (int ch = 0; ch < nch; ++ch) {
    const int kbase = ch << 9;
    // stage 16x512 A chunk into LDS (float4, coalesced-ish)
    for (int idx = tid * 4; idx < 16 * 512; idx += 1024) {
      const int m  = idx >> 9;
      const int kk = idx & 511;
      const int kg = kbase + kk;
      const float* src;
      if (layer == 0 && kg < Ka) {
        const int tok = (t == 0) ? 1 : tokens[m * SEQL + (t - 1)];   // SOS=1
        src = emb + (size_t)tok * EE + kg;
      } else if (kg < Ka) {
        src = h_below + m * HH + kg;
      } else {
        src = h_prev + m * HH + (kg - Ka);
      }
      *(float4*)&As[idx] = *(const float4*)src;
    }
    __syncthreads();

    const float* wrow;                   // per-lane weight row, k-contiguous
    if (kbase < Ka) wrow = W_ih + (size_t)(gcol + mn) * ih_stride + kbase;
    else            wrow = W_hh + (size_t)(gcol + mn) * HH + (kbase - Ka);

    // fold the wave's K-half into the base pointers -> literal loop bounds
    const float* arowK = &As[mn * 512 + kstart];
    const float* wrowK = wrow + kstart;
#pragma unroll 4
    for (int k0 = 0; k0 < 256; k0 += 8) {
      const int ka = k0 + 2 * kh;
      const int kb = ka + 4;
      v2f a0 = {arowK[ka], arowK[ka + 1]};
      v2f b0 = {wrowK[ka], wrowK[ka + 1]};
      acc0 = wmma4(a0, b0, acc0);
      v2f a1 = {arowK[kb], arowK[kb + 1]};
      v2f b1 = {wrowK[kb], wrowK[kb + 1]};
      acc1 = wmma4(a1, b1, acc1);
    }
    __syncthreads();
  }

  // write the eight partial 16x16 gate tiles through LDS
#pragma unroll
  for (int r = 0; r < 8; ++r)
    Gt[wave * 256 + (r + 8 * kh) * 16 + mn] = acc0[r] + acc1[r];
  __syncthreads();

  // pointwise LSTM update for this block's 16 h-columns (one elem per thread)
  {
    const int e   = tid;                 // 0..255
    const int m   = e >> 4;
    const int col = n0 + (e & 15);
    const float ig = Gt[0 * 256 + e] + Gt[4 * 256 + e] + b_ih[col]          + b_hh[col];
    const float fg = Gt[1 * 256 + e] + Gt[5 * 256 + e] + b_ih[HH + col]     + b_hh[HH + col];
    const float gg = Gt[2 * 256 + e] + Gt[6 * 256 + e] + b_ih[2 * HH + col] + b_hh[2 * HH + col];
    const float og = Gt[3 * 256 + e] + Gt[7 * 256 + e] + b_ih[3 * HH + col] + b_hh[3 * HH + col];
    const float i = sigmoidf_(ig);
    const float f = sigmoidf_(fg);
    const float g = tanhf(gg);
    const float o = sigmoidf_(og);
    const float c = f * c_state[m * HH + col] + i * g;
    const float h = o * tanhf(c);
    c_state[m * HH + col] = c;
    h_out[m * HH + col]   = h;
    if (hs) hs[((size_t)m * TT + t) * HH + col] = h;
  }
}

// ---------------------------------------------------------------------------
// scores[b,t,s] = hs[b,t,:] . enc[b,s,:]   (K = 1024)
// grid = (SS/64, 17, B), block = 128 (4 waves, one 16x16 n-tile each)
// ---------------------------------------------------------------------------
__global__ __launch_bounds__(128) void attn_scores_kernel(
    const float* __restrict__ hs, const float* __restrict__ enc,
    float* __restrict__ sc)
{
  const int tid = threadIdx.x, wave = tid >> 5, lane = tid & 31;
  const int kh = lane >> 4, mn = lane & 15;
  const int b = blockIdx.z, mt = blockIdx.y;
  const int n0 = (blockIdx.x * 4 + wave) * 16;
  int mr = mt * 16 + mn; if (mr > TT - 1) mr = TT - 1;   // clamp pad rows
  const float* aP = hs  + ((size_t)b * TT + mr) * HH;
  const float* bP = enc + ((size_t)b * SS + n0 + mn) * HH;
  v8f acc = {0.f, 0.f, 0.f, 0.f, 0.f, 0.f, 0.f, 0.f};
  acc = gemm_tile(aP, 1, bP, 1, HH, kh, acc);
#pragma unroll
  for (int r = 0; r < 8; ++r) {
    const int m = mt * 16 + r + 8 * kh;
    if (m < TT) sc[((size_t)b * TT + m) * SS + n0 + mn] = acc[r];
  }
}

// in-place softmax over S=512 per (b,t) row
__global__ __launch_bounds__(128) void softmax_kernel(float* __restrict__ sc) {
  __shared__ float red[128];
  float* p = sc + (size_t)blockIdx.x * SS;
  const int tid = threadIdx.x;
  float mx = -1e30f;
  for (int i = tid; i < SS; i += 128) mx = fmaxf(mx, p[i]);
  red[tid] = mx; __syncthreads();
  for (int s = 64; s; s >>= 1) { if (tid < s) red[tid] = fmaxf(red[tid], red[tid + s]); __syncthreads(); }
  mx = red[0]; __syncthreads();
  float sum = 0.f;
  for (int i = tid; i < SS; i += 128) { const float e = __expf(p[i] - mx); p[i] = e; sum += e; }
  red[tid] = sum; __syncthreads();
  for (int s = 64; s; s >>= 1) { if (tid < s) red[tid] += red[tid + s]; __syncthreads(); }
  const float inv = 1.f / red[0];
  __syncthreads();
  for (int i = tid; i < SS; i += 128) p[i] *= inv;
}

// ctx[b,t,h] = sum_s attn[b,t,s] * enc[b,s,h]   (K = 512, B-stride = HH)
__global__ __launch_bounds__(128) void attn_ctx_kernel(
    const float* __restrict__ at, const float* __restrict__ enc,
    float* __restrict__ ctx)
{
  const int tid = threadIdx.x, wave = tid >> 5, lane = tid & 31;
  const int kh = lane >> 4, mn = lane & 15;
  const int b = blockIdx.z, mt = blockIdx.y;
  const int n0 = (blockIdx.x * 4 + wave) * 16;
  int mr = mt * 16 + mn; if (mr > TT - 1) mr = TT - 1;
  const float* aP = at  + ((size_t)b * TT + mr) * SS;
  const float* bP = enc + (size_t)b * SS * HH + (n0 + mn);
  v8f acc = {0.f, 0.f, 0.f, 0.f, 0.f, 0.f, 0.f, 0.f};
  acc = gemm_tile(aP, 1, bP, HH, SS, kh, acc);
#pragma unroll
  for (int r = 0; r < 8; ++r) {
    const int m = mt * 16 + r + 8 * kh;
    if (m < TT) ctx[((size_t)b * TT + m) * HH + n0 + mn] = acc[r];
  }
}

// hidden = tanh([hs | ctx] @ W1^T + b1); flat M = B*T = 4112 = 257*16 rows
__global__ __launch_bounds__(128) void mlp1_kernel(
    const float* __restrict__ hs, const float* __restrict__ ctx,
    const float* __restrict__ W1, const float* __restrict__ b1,
    float* __restrict__ hid)
{
  const int tid = threadIdx.x, wave = tid >> 5, lane = tid & 31;
  const int kh = lane >> 4, mn = lane & 15;
  const int mrow = blockIdx.y * 16 + mn;
  const int n0 = (blockIdx.x * 4 + wave) * 16;
  const float* w = W1 + (size_t)(n0 + mn) * (2 * HH);
  v8f acc = {0.f, 0.f, 0.f, 0.f, 0.f, 0.f, 0.f, 0.f};
  acc = gemm_tile(hs  + (size_t)mrow * HH, 1, w,      1, HH, kh, acc);
  acc = gemm_tile(ctx + (size_t)mrow * HH, 1, w + HH, 1, HH, kh, acc);
#pragma unroll
  for (int r = 0; r < 8; ++r) {
    const int m = blockIdx.y * 16 + r + 8 * kh;
    hid[(size_t)m * HH + n0 + mn] = tanhf(acc[r] + b1[n0 + mn]);
  }
}

// logits = hidden @ W2^T + b2
__global__ __launch_bounds__(128) void mlp2_kernel(
    const float* __restrict__ hid, const float* __restrict__ W2,
    const float* __restrict__ b2, float* __restrict__ lg)
{
  const int tid = threadIdx.x, wave = tid >> 5, lane = tid & 31;
  const int kh = lane >> 4, mn = lane & 15;
  const int mrow = blockIdx.y * 16 + mn;
  const int n0 = (blockIdx.x * 4 + wave) * 16;
  const float* w = W2 + (size_t)(n0 + mn) * HH;
  v8f acc = {0.f, 0.f, 0.f, 0.f, 0.f, 0.f, 0.f, 0.f};
  acc = gemm_tile(hid + (size_t)mrow * HH, 1, w, 1, HH, kh, acc);
#pragma unroll
  for (int r = 0; r < 8; ++r) {
    const int m = blockIdx.y * 16 + r + 8 * kh;
    lg[(size_t)m * VV + n0 + mn] = acc[r] + b2[n0 + mn];
  }
}

// per-(b,t) NLL = max + log(sum exp(x-max)) - logit[target]
__global__ __launch_bounds__(128) void loss_kernel(
    const float* __restrict__ lg, const int* __restrict__ tokens,
    float* __restrict__ nll)
{
  __shared__ float red[128];
  const int row = blockIdx.x;              // b*TT + t
  const int b = row / TT, t = row % TT;
  const float* p = lg + (size_t)row * VV;
  const int tid = threadIdx.x;
  float mx = -1e30f;
  for (int i = tid; i < VV; i += 128) mx = fmaxf(mx, p[i]);
  red[tid] = mx; __syncthreads();
  for (int s = 64; s; s >>= 1) { if (tid < s) red[tid] = fmaxf(red[tid], red[tid + s]); __syncthreads(); }
  mx = red[0]; __syncthreads();
  float sum = 0.f;
  for (int i = tid; i < VV; i += 128) sum += __expf(p[i] - mx);
  red[tid] = sum; __syncthreads();
  for (int s = 64; s; s >>= 1) { if (tid < s) red[tid] += red[tid + s]; __syncthreads(); }
  if (tid == 0) {
    const int tgt = (t < SEQL) ? tokens[b * SEQL + t] : 2;   // EOS=2
    nll[row] = mx + __logf(red[0]) - p[tgt];
  }
}

__global__ void reduce_kernel(const float* __restrict__ nll, float* __restrict__ out) {
  __shared__ float red[256];
  const int tid = threadIdx.x;
  float s = 0.f;
  for (int i = tid; i < BATCH * TT; i += 256) s += nll[i];
  red[tid] = s; __syncthreads();
  for (int k = 128; k; k >>= 1) { if (tid < k) red[tid] += red[tid + k]; __syncthreads(); }
  if (tid == 0) out[0] = red[0] / (float)(BATCH * TT);
}

__global__ void zero_kernel(float* __restrict__ p, int n) {
  const int i = blockIdx.x * 256 + threadIdx.x;
  if (i < n) p[i] = 0.f;
}

// ---------------------------------------------------------------------------
extern "C" void kernel_launch(void* const* d_in, const int* in_sizes, int n_in,
                              void* d_out, int out_size, void* d_ws, size_t ws_size,
                              hipStream_t stream) {
  (void)in_sizes; (void)n_in; (void)out_size; (void)ws_size;
  const int*   tokens = (const int*)d_in[0];
  const float* enc    = (const float*)d_in[1];
  const float* emb    = (const float*)d_in[2];
  const float* W_ih[3] = {(const float*)d_in[3], (const float*)d_in[7],  (const float*)d_in[11]};
  const float* W_hh[3] = {(const float*)d_in[4], (const float*)d_in[8],  (const float*)d_in[12]};
  const float* b_ih[3] = {(const float*)d_in[5], (const float*)d_in[9],  (const float*)d_in[13]};
  const float* b_hh[3] = {(const float*)d_in[6], (const float*)d_in[10], (const float*)d_in[14]};
  const float* W1 = (const float*)d_in[15];
  const float* b1 = (const float*)d_in[16];
  const float* W2 = (const float*)d_in[17];
  const float* b2 = (const float*)d_in[18];

  float* w = (float*)d_ws;
  const size_t BHf = (size_t)BATCH * HH;                 // 16384
  float* st  = w;                                        // 9*BHf: per layer {hA,hB,c}
  float* hsb = st  + 9 * BHf;                            // [B][T][H]
  float* scb = hsb + (size_t)BATCH * TT * HH;            // [B][T][S]
  float* ctb = scb + (size_t)BATCH * TT * SS;            // [B][T][H]
  float* hid = ctb + (size_t)BATCH * TT * HH;            // [B*T][H]
  float* lgb = hid + (size_t)BATCH * TT * HH;            // [B*T][V]
  float* nll = lgb + (size_t)BATCH * TT * VV;            // [B*T]

  zero_kernel<<<(9 * (int)BHf + 255) / 256, 256, 0, stream>>>(st, 9 * (int)BHf);

  // sequential recurrence: 257 steps x 3 layers (stream-ordered launches)
  for (int t = 0; t < TT; ++t) {
    const int pr = t & 1;
    float* hp[3]; float* ho[3]; float* cs[3];
    for (int L = 0; L < 3; ++L) {
      hp[L] = st + (size_t)L * 3 * BHf + (size_t)pr * BHf;
      ho[L] = st + (size_t)L * 3 * BHf + (size_t)(pr ^ 1) * BHf;
      cs[L] = st + (size_t)L * 3 * BHf + 2 * BHf;
    }
    lstm_step_kernel<<<64, 256, 0, stream>>>(t, 0, tokens, emb,
        W_ih[0], EE + HH, EE, W_hh[0], b_ih[0], b_hh[0],
        (const float*)nullptr, hp[0], ho[0], cs[0], (float*)nullptr);
    lstm_step_kernel<<<64, 256, 0, stream>>>(t, 1, tokens, emb,
        W_ih[1], HH, HH, W_hh[1], b_ih[1], b_hh[1],
        ho[0], hp[1], ho[1], cs[1], (float*)nullptr);
    lstm_step_kernel<<<64, 256, 0, stream>>>(t, 2, tokens, emb,
        W_ih[2], HH, HH, W_hh[2], b_ih[2], b_hh[2],
        ho[1], hp[2], ho[2], cs[2], hsb);
  }

  attn_scores_kernel<<<dim3(SS / 64, 17, BATCH), 128, 0, stream>>>(hsb, enc, scb);
  softmax_kernel<<<BATCH * TT, 128, 0, stream>>>(scb);
  attn_ctx_kernel<<<dim3(HH / 64, 17, BATCH), 128, 0, stream>>>(scb, enc, ctb);
  mlp1_kernel<<<dim3(HH / 64, TT), 128, 0, stream>>>(hsb, ctb, W1, b1, hid);
  mlp2_kernel<<<dim3(VV / 64, TT), 128, 0, stream>>>(hid, W2, b2, lgb);
  loss_kernel<<<BATCH * TT, 128, 0, stream>>>(lgb, tokens, nll);
  reduce_kernel<<<1, 256, 0, stream>>>(nll, (float*)d_out);
}